// VAELoss_64235530879334
// MI455X (gfx1250) — compile-verified
//
#include <hip/hip_runtime.h>
#include <hip/hip_bf16.h>
#include <math.h>

typedef float v2f __attribute__((ext_vector_type(2)));
typedef float v8f __attribute__((ext_vector_type(8)));

#define EPS_F 1e-7f

// One wave computes, for a 16-row tile of P, the min over all M columns of Q
// of the pairwise L2 distance, via V_WMMA_F32_16X16X4_F32.
//
// Algebraic fold: A row = (px,py,pz, -|p|^2/2), B col = (qx,qy,qz, 1), so the
// WMMA yields c = p.q - |p|^2/2 and d^2 = |q|^2 - 2c. The running min is kept
// in squared-distance space; sqrt(max(.,1e-12)) is monotonic and applied once
// per row at the end.
__global__ __launch_bounds__(256) void chamfer_rowmin_kernel(
    const float* __restrict__ P,   // (bs, N, 3) row points
    const float* __restrict__ Q,   // (bs, M, 3) col points
    float* __restrict__ outmin,    // (bs, N) per-row nearest-neighbor distance
    int N, int M)
{
  const int tilesPerBatch = N / 128;
  const int b    = blockIdx.x / tilesPerBatch;
  const int tile = blockIdx.x % tilesPerBatch;
  const int wave = threadIdx.x >> 5;
  const int lane = threadIdx.x & 31;
  const int hi   = lane >> 4;     // 0 = K comps {0,1}, 1 = K comps {2,3}
  const int c16  = lane & 15;     // row (A) / col (B,C) index within 16-tile

  const int rowbase = tile * 128 + wave * 16;
  const float* Pb = P + (size_t)b * N * 3;
  const float* Qb = Q + (size_t)b * M * 3;

  // ---- A fragment: constant across the whole M sweep ----
  // 32-bit A 16x4 layout: lanes 0-15 -> rows 0-15 K={0,1}; lanes 16-31 -> K={2,3}.
  // K=3 slot carries -|p|^2/2.
  v2f afrag;
  {
    const float* pr = Pb + (size_t)(rowbase + c16) * 3;
    float px = pr[0], py = pr[1], pz = pr[2];
    float n1 = px * px + py * py + pz * pz;
    afrag.x = hi ? pz : px;
    afrag.y = hi ? (-0.5f * n1) : py;
  }

  float rmin[8];                  // running min of SQUARED distance
#pragma unroll
  for (int v = 0; v < 8; ++v) rmin[v] = 3.4e38f;

  // ---- sweep columns: two 16-wide tiles per iteration for WMMA ILP ----
  for (int mt = 0; mt < M; mt += 32) {
    const float* q0 = Qb + (size_t)(mt + c16) * 3;
    float q0x = q0[0], q0y = q0[1], q0z = q0[2];
    float n2a = q0x * q0x + q0y * q0y + q0z * q0z;
    v2f bf0; bf0.x = hi ? q0z : q0x; bf0.y = hi ? 1.0f : q0y;

    const float* q1 = Qb + (size_t)(mt + 16 + c16) * 3;
    float q1x = q1[0], q1y = q1[1], q1z = q1[2];
    float n2b = q1x * q1x + q1y * q1y + q1z * q1z;
    v2f bf1; bf1.x = hi ? q1z : q1x; bf1.y = hi ? 1.0f : q1y;

    v8f c0 = {0.f, 0.f, 0.f, 0.f, 0.f, 0.f, 0.f, 0.f};
    v8f c1 = {0.f, 0.f, 0.f, 0.f, 0.f, 0.f, 0.f, 0.f};
    c0 = __builtin_amdgcn_wmma_f32_16x16x4_f32(false, afrag, false, bf0,
                                               (short)0, c0, false, false);
    c1 = __builtin_amdgcn_wmma_f32_16x16x4_f32(false, afrag, false, bf1,
                                               (short)0, c1, false, false);

#pragma unroll
    for (int v = 0; v < 8; ++v) {
      float d2a = n2a - 2.0f * c0[v];   // = |p|^2 + |q|^2 - 2 p.q
      float d2b = n2b - 2.0f * c1[v];
      rmin[v] = fminf(rmin[v], fminf(d2a, d2b));
    }
  }

  // ---- butterfly min across the 16 lanes that partition the columns,
  //      then the single sqrt per row (monotonic, hoisted out of the loop) ----
#pragma unroll
  for (int v = 0; v < 8; ++v) {
    float m = rmin[v];
    m = fminf(m, __shfl_xor(m, 1, 32));
    m = fminf(m, __shfl_xor(m, 2, 32));
    m = fminf(m, __shfl_xor(m, 4, 32));
    m = fminf(m, __shfl_xor(m, 8, 32));
    rmin[v] = __builtin_sqrtf(fmaxf(m, 1e-12f));
  }
  if (c16 == 0) {
#pragma unroll
    for (int v = 0; v < 8; ++v)
      outmin[(size_t)b * N + rowbase + 8 * hi + v] = rmin[v];
  }
}

// Single-block final reduction: sum(chamfer mins) + mean BCE + CR.
__global__ __launch_bounds__(256) void reduce_loss_kernel(
    const float* __restrict__ prob_pred,
    const float* __restrict__ prob_target,
    const float* __restrict__ mu,
    const float* __restrict__ lb,
    const float* __restrict__ ub,
    const float* __restrict__ mins, int nmins,
    int bsK, int K, float* __restrict__ out)
{
  __shared__ float red[256];
  const int t = threadIdx.x;
  float acc = 0.0f;

  for (int i = t; i < nmins; i += 256) acc += mins[i];

  float bce = 0.0f;
  for (int i = t; i < bsK; i += 256) {
    float p  = fminf(fmaxf(prob_pred[i], EPS_F), 1.0f - EPS_F);
    float tt = prob_target[i];
    bce -= tt * logf(p) + (1.0f - tt) * log1pf(-p);
  }
  acc += bce / (float)bsK;

  for (int i = t; i < K; i += 256) {
    float cr = fabsf(mu[i]) - 0.5f * (lb[i] + ub[i]) + 1.0f / (float)K;
    acc += fmaxf(cr, 0.0f);
  }

  red[t] = acc;
  __syncthreads();
  for (int s = 128; s > 0; s >>= 1) {
    if (t < s) red[t] += red[t + s];
    __syncthreads();
  }
  if (t == 0) out[0] = red[0];
}

extern "C" void kernel_launch(void* const* d_in, const int* in_sizes, int n_in,
                              void* d_out, int out_size, void* d_ws, size_t ws_size,
                              hipStream_t stream) {
  const float* prob_pred   = (const float*)d_in[0];
  const float* prob_target = (const float*)d_in[1];
  const float* P           = (const float*)d_in[2];  // x_diff_pred  (bs,N,3)
  const float* Qm          = (const float*)d_in[3];  // x_diff_target(bs,M,3)
  const float* mu          = (const float*)d_in[4];
  const float* lb          = (const float*)d_in[5];
  const float* ub          = (const float*)d_in[6];
  float* out = (float*)d_out;

  const int K   = in_sizes[4];
  const int bsK = in_sizes[0];
  const int bs  = bsK / K;
  const int N   = in_sizes[2] / (bs * 3);
  const int M   = in_sizes[3] / (bs * 3);

  float* ws   = (float*)d_ws;
  float* minP = ws;                      // (bs, N) nearest dist p1 -> p2
  float* minQ = ws + (size_t)bs * N;     // (bs, M) nearest dist p2 -> p1

  // Pass 1: rows = p1, cols = p2.  Pass 2: swapped (avoids cross-WG atomics).
  chamfer_rowmin_kernel<<<bs * (N / 128), 256, 0, stream>>>(P,  Qm, minP, N, M);
  chamfer_rowmin_kernel<<<bs * (M / 128), 256, 0, stream>>>(Qm, P,  minQ, M, N);

  reduce_loss_kernel<<<1, 256, 0, stream>>>(prob_pred, prob_target, mu, lb, ub,
                                            ws, bs * (N + M), bsK, K, out);
}